// ExtrinsicOptimizer_38732015076060
// MI455X (gfx1250) — compile-verified
//
#include <hip/hip_runtime.h>

typedef __attribute__((ext_vector_type(2))) float v2f;
typedef __attribute__((ext_vector_type(4))) float v4f;
typedef __attribute__((ext_vector_type(8))) float v8f;

#define SE3_EPS 1e-4f

// One block = 256 threads = 8 waves; each wave owns 32 consecutive elements.
// Phase 1: lane i builds the 4x4 SE3-exp refinement matrix for element base+i in LDS.
// Phase 2: 8x V_WMMA_F32_16X16X4_F32 per wave. Operands are swapped vs the naive
//          mapping (A = 4 stacked extrinsic-transposes from global, B = 4 refinement
//          transposes from LDS) so the diagonal 4x4 blocks of D come out transposed:
//          each useful lane then holds one contiguous ROW of an output matrix and
//          writes it with a single ds_store_b128 (branch-free via address select).
// Store:   staged row-major in LDS -> contiguous 512B-per-instruction global stores.
__global__ __launch_bounds__(256, 2) void se3_refine_wmma(
    const float* __restrict__ extr,      // [N,4,4] row-major
    const float* __restrict__ se3w,      // [Nv,6]
    const int*   __restrict__ indices,   // [N]
    float*       __restrict__ out,       // [N,4,4]
    int N)
{
  // 16 KB staging (one 4x4 per element) + 4 KB dump area for inactive-lane stores.
  __shared__ __align__(16) float lds[256 * 16 + 256 * 4];
  const int lane = threadIdx.x & 31;
  const int wave = threadIdx.x >> 5;
  float* wbuf = lds + wave * (32 * 16);
  float* dumpSlot = lds + 256 * 16 + threadIdx.x * 4;

  const int base = blockIdx.x * 256 + wave * 32;  // first element of this wave
  const int e    = base + lane;
  const int ec   = (e < N) ? e : (N - 1);
  const bool fullWave = (base + 32 <= N);         // wave-uniform

  // Warm caches for the extrinsics this wave consumes in phase 2.
  __builtin_prefetch(extr + (long long)ec * 16, 0, 1);

  // ---------------- Phase 1: per-lane SE(3) exp map -> LDS -----------------
  {
    const int j = indices[ec];
    const float* wp = se3w + (long long)j * 6;       // 8-byte aligned rows
    v2f w01 = *(const v2f*)(wp + 0);
    v2f w23 = *(const v2f*)(wp + 2);
    v2f w45 = *(const v2f*)(wp + 4);
    const float tx = w01.x, ty = w01.y, tz = w23.x;
    const float rx = w23.y, ry = w45.x, rz = w45.y;

    const float dot = rx*rx + ry*ry + rz*rz;         // unclamped, used in K^2
    const float th  = sqrtf(fmaxf(dot, SE3_EPS));    // clamped theta
    const float inv = 1.0f / th;
    const float s = __sinf(th), c = __cosf(th);
    const float A  = s * inv;
    const float Bb = (1.0f - c) * inv * inv;
    const float Cc = (th - s) * inv * inv * inv;

    // hat(r)^2 = r r^T - |r|^2 I
    const float xx = rx*rx - dot, yy = ry*ry - dot, zz = rz*rz - dot;
    const float xy = rx*ry, xz = rx*rz, yz = ry*rz;

    const float R00 = 1.0f + Bb*xx,  R01 = -A*rz + Bb*xy, R02 =  A*ry + Bb*xz;
    const float R10 =  A*rz + Bb*xy, R11 = 1.0f + Bb*yy,  R12 = -A*rx + Bb*yz;
    const float R20 = -A*ry + Bb*xz, R21 =  A*rx + Bb*yz, R22 = 1.0f + Bb*zz;

    const float V00 = 1.0f + Cc*xx,   V01 = -Bb*rz + Cc*xy, V02 =  Bb*ry + Cc*xz;
    const float V10 =  Bb*rz + Cc*xy, V11 = 1.0f + Cc*yy,   V12 = -Bb*rx + Cc*yz;
    const float V20 = -Bb*ry + Cc*xz, V21 =  Bb*rx + Cc*yz, V22 = 1.0f + Cc*zz;

    const float Tx = V00*tx + V01*ty + V02*tz;
    const float Ty = V10*tx + V11*ty + V12*tz;
    const float Tz = V20*tx + V21*ty + V22*tz;

    v4f* m = (v4f*)(wbuf + lane * 16);
    m[0] = (v4f){R00, R01, R02, Tx};
    m[1] = (v4f){R10, R11, R12, Ty};
    m[2] = (v4f){R20, R21, R22, Tz};
    m[3] = (v4f){0.f, 0.f, 0.f, 1.f};
  }
  __syncthreads();

  // ---------------- Phase 2: WMMA 16x16x4 f32, 4 elements per op ----------
  const int m16 = lane & 15;
  const int hi  = lane >> 4;            // 0 -> K{0,1}, 1 -> K{2,3}
  const int k0  = hi << 1;
  const int sub = m16 >> 2;             // which of the 4 matrices in the group
  const int rc  = m16 & 3;              // col-in-E for A, row-in-R for B
  const bool useful = ((sub >> 1) == hi);
  const bool pickHi = (sub & 1) != 0;

  // D = A x B with A[m][k] = E_{m/4}[k][m%4], B[k][n] = R_{n/4}[n%4][k]
  //   => diagonal block i of D = (R_i @ E_i)^T, so a useful lane's 4 values are
  //      one contiguous row (rc) of output matrix `sub`.
  auto group = [&](int g, const float* aPtr) {
    v2f b = *(const v2f*)(wbuf + (g*4 + sub)*16 + rc*4 + k0);   // ds_load_b64
    v2f a; a.x = aPtr[0]; a.y = aPtr[4];                        // 2x global b32
    v8f cz = {};
    v8f d = __builtin_amdgcn_wmma_f32_16x16x4_f32(
        /*neg_a=*/false, a, /*neg_b=*/false, b,
        /*c_mod=*/(short)0, cz, /*reuse_a=*/false, /*reuse_b=*/false);
    v4f val;                                 // branch-free half select
    val.x = pickHi ? d[4] : d[0];
    val.y = pickHi ? d[5] : d[1];
    val.z = pickHi ? d[6] : d[2];
    val.w = pickHi ? d[7] : d[3];
    float* dst = useful ? (wbuf + (g*4 + sub)*16 + rc*4) : dumpSlot;
    *(v4f*)dst = val;                        // ds_store_b128, no exec games
  };

  const float* gA = extr + (long long)(base + sub) * 16 + k0*4 + rc;
  if (fullWave) {
    #pragma unroll
    for (int g = 0; g < 8; ++g) group(g, gA + g * 64);  // immediate offsets
  } else {
    #pragma unroll
    for (int g = 0; g < 8; ++g) {
      const int eb  = base + g*4 + sub;
      const int ebc = (eb < N) ? eb : (N - 1);
      group(g, extr + (long long)ebc * 16 + k0*4 + rc);
    }
  }
  __syncthreads();

  // ---------------- Coalesced output: 4x b128 per lane, 512B/instr --------
  const long long outBase = (long long)base * 16;   // in floats
  if (fullWave) {
    #pragma unroll
    for (int q = 0; q < 4; ++q) {
      const int f = q * 128 + lane * 4;
      *(v4f*)(out + outBase + f) = *(const v4f*)(wbuf + f);
    }
  } else {
    const long long limit = (long long)N * 16;
    #pragma unroll
    for (int q = 0; q < 4; ++q) {
      const int f = q * 128 + lane * 4;
      const long long fg = outBase + f;
      if (fg < limit) {
        *(v4f*)(out + fg) = *(const v4f*)(wbuf + f);
      }
    }
  }
}

extern "C" void kernel_launch(void* const* d_in, const int* in_sizes, int n_in,
                              void* d_out, int out_size, void* d_ws, size_t ws_size,
                              hipStream_t stream) {
  (void)n_in; (void)out_size; (void)d_ws; (void)ws_size;
  const float* extr = (const float*)d_in[0];   // extrinsics [B,4,4] f32
  const float* se3w = (const float*)d_in[1];   // se3_weight [Nv,6] f32
  const int*   idx  = (const int*)d_in[2];     // indices [B] i32
  float* out = (float*)d_out;
  const int N = in_sizes[2];
  if (N <= 0) return;
  const int blocks = (N + 255) / 256;
  se3_refine_wmma<<<blocks, 256, 0, stream>>>(extr, se3w, idx, out, N);
}